// Up_3599182594618
// MI455X (gfx1250) — compile-verified
//
#include <hip/hip_runtime.h>
#include <hip/hip_bf16.h>

// ---------------------------------------------------------------------------
// Problem constants (static shapes from setup_inputs)
// ---------------------------------------------------------------------------
#define BATCH 2
#define D_OUT 16
#define H_OUT 32
#define W_OUT 32
#define SPAT  (D_OUT * H_OUT * W_OUT)      // 16384 spatial positions per batch
#define NTOK  (BATCH * SPAT)               // 32768 tokens
#define CMODEL 32
#define DINNER 64
#define DSTATE 16
#define DTRANK 2
#define XPROJ_N (DTRANK + 2 * DSTATE)      // 34

#define NCHUNK 128
#define CLEN   (SPAT / NCHUNK)             // 128 steps per chunk

// workspace layout (float offsets)
#define WS_XCAT 0L                               // [2][64][16384]   2097152
#define WS_T1   2097152L                         // [2][32][16384]   1048576
#define WS_T2   3145728L                         // [2][32][16384]   1048576
#define WS_T3   4194304L                         // [2][32][16384]   1048576
#define WS_TOK  5242880L                         // [32768][32]      1048576
#define WS_XZ   6291456L                         // [32768][128]     4194304
#define WS_XIA  0L                               // reuse XCAT: [32768][64] 2097152
#define WS_XDBL 2097152L                         // reuse T1/T2: [32768][34] 1114112
#define WS_P    3211264L                         // [2][128][64][16] 262144 (hole after XDBL)
#define WS_Q    3473408L                         // [2][128][64][16] 262144
#define WS_H0   3735552L                         // [2][128][64][16] 262144 (ends 3997696 < T3)
#define WS_DT   4194304L                         // reuse T3/TOK: [32768][64] 2097152
#define WS_YS   10485760L                        // [32768][64]      2097152
// total = 12582912 floats = 48 MiB

typedef float v2f __attribute__((ext_vector_type(2)));
typedef float v8f __attribute__((ext_vector_type(8)));

__device__ __forceinline__ float silu_f(float x) { return x / (1.0f + __expf(-x)); }
__device__ __forceinline__ float leaky_f(float x) { return x >= 0.0f ? x : 0.01f * x; }
__device__ __forceinline__ float softplus_f(float x) {
    return x > 20.0f ? x : log1pf(__expf(x));
}

// ---------------------------------------------------------------------------
// Generic strided fp32 GEMM on the WMMA pipe: C = A @ W^T (+bias)
//   A: [M,K] with strides (sAM, sAK) per batch slice (aBatch per blockIdx.z)
//   W: [N,K] row-major (shared across batch) -> B[k][n] = W[n][k]
//   C: [M,N] with strides (sCM, sCN), cBatch per blockIdx.z
// One wave (32 threads) computes one 16x16 tile via V_WMMA_F32_16X16X4_F32
// (full fp32 A/B -> zero precision loss vs the fp32 reference).
// Partial N tiles: clamp W row on load (EXEC must stay all-ones for WMMA),
// guard on store only.
// ---------------------------------------------------------------------------
__global__ __launch_bounds__(32) void gemm_wmma_f32(
    const float* __restrict__ A, const float* __restrict__ W,
    const float* __restrict__ bias, float* __restrict__ C,
    int M, int N, int K,
    long aBatch, long sAM, long sAK,
    long cBatch, long sCM, long sCN)
{
    const int lane = threadIdx.x;    // 0..31
    const int half = lane >> 4;      // 0: K 0-1 / M 0-7 rows of acc, 1: K 2-3 / M 8-15
    const int idx  = lane & 15;

    const long m0 = (long)blockIdx.x * 16;
    const int  n0 = blockIdx.y * 16;
    const long abase = (long)blockIdx.z * aBatch;
    const long cbase = (long)blockIdx.z * cBatch;

    // A fragment row: m = m0 + idx (same for both halves; halves split K)
    const long arow = abase + (m0 + idx) * sAM;
    // B fragment column: n = n0 + idx  -> W row (clamped for partial tiles)
    int wrow = n0 + idx;
    if (wrow >= N) wrow = N - 1;
    const float* __restrict__ wr = W + (long)wrow * K;

    v8f acc = {};
    for (int k0 = 0; k0 < K; k0 += 4) {
        const int ka = k0 + 2 * half;
        v2f va, vb;
        va.x = A[arow + (long)(ka + 0) * sAK];
        va.y = A[arow + (long)(ka + 1) * sAK];
        vb.x = wr[ka + 0];
        vb.y = wr[ka + 1];
        acc = __builtin_amdgcn_wmma_f32_16x16x4_f32(
            /*neg_a=*/false, va, /*neg_b=*/false, vb,
            /*c_mod=*/(short)0, acc, /*reuse_a=*/false, /*reuse_b=*/false);
    }

    const int n = n0 + idx;
    if (n < N) {
        const float badd = bias ? bias[n] : 0.0f;
#pragma unroll
        for (int r = 0; r < 8; ++r) {
            const int m = r + 8 * half;           // C layout: VGPR r -> M=r (+8 for hi half)
            C[cbase + (m0 + m) * sCM + (long)n * sCN] = acc[r] + badd;
        }
    }
}

// ---------------------------------------------------------------------------
// ConvTranspose3d(k=2,s=2) of x1 into channels [32,64) of xcat; copy x2 into
// channels [0,32).  xcat layout: [n][c][s] with s over (z,y,x) of 16x32x32.
// ---------------------------------------------------------------------------
__global__ void upsample_concat_kernel(
    const float* __restrict__ x1, const float* __restrict__ x2,
    const float* __restrict__ upw, const float* __restrict__ upb,
    float* __restrict__ xcat)
{
    const int i = blockIdx.x * blockDim.x + threadIdx.x;
    const int total = BATCH * 64 * SPAT;
    if (i >= total) return;
    const int s = i % SPAT;
    const int c = (i / SPAT) % 64;
    const int n = i / (SPAT * 64);
    if (c < 32) { xcat[i] = x2[((long)(n * 32 + c)) * SPAT + s]; return; }
    const int o = c - 32;
    const int xw = s % W_OUT;
    const int yw = (s / W_OUT) % H_OUT;
    const int zw = s / (W_OUT * H_OUT);
    const int zi = zw >> 1, yi = yw >> 1, xi = xw >> 1;
    const int di = zw & 1, hj = yw & 1, wk = xw & 1;
    float acc = upb[o];
    // x1: [2][32][8][16][16]; up_w: [32][32][2][2][2] (c,o,i,j,k)
#pragma unroll 4
    for (int c2 = 0; c2 < 32; ++c2) {
        const float xv = x1[(((long)(n * 32 + c2) * 8 + zi) * 16 + yi) * 16 + xi];
        const float wv = upw[((((long)c2 * 32 + o) * 2 + di) * 2 + hj) * 2 + wk];
        acc = fmaf(xv, wv, acc);
    }
    xcat[i] = acc;
}

// ---------------------------------------------------------------------------
// InstanceNorm3d(affine=False) + LeakyReLU, in place.  One block per (n,c).
// ---------------------------------------------------------------------------
__global__ __launch_bounds__(256) void instnorm_leaky_kernel(float* __restrict__ x)
{
    float* p = x + (long)blockIdx.x * SPAT;
    __shared__ float ssum[256], ssq[256];
    float s = 0.0f, q = 0.0f;
    for (int i = threadIdx.x; i < SPAT; i += 256) {
        const float v = p[i];
        s += v; q = fmaf(v, v, q);
    }
    ssum[threadIdx.x] = s; ssq[threadIdx.x] = q;
    __syncthreads();
    for (int off = 128; off > 0; off >>= 1) {
        if (threadIdx.x < off) {
            ssum[threadIdx.x] += ssum[threadIdx.x + off];
            ssq[threadIdx.x]  += ssq[threadIdx.x + off];
        }
        __syncthreads();
    }
    const float mean = ssum[0] * (1.0f / SPAT);
    const float var  = ssq[0] * (1.0f / SPAT) - mean * mean;
    const float inv  = rsqrtf(var + 1e-5f);
    for (int i = threadIdx.x; i < SPAT; i += 256)
        p[i] = leaky_f((p[i] - mean) * inv);
}

// ---------------------------------------------------------------------------
// Grouped 3x3x3 conv (groups=8, 4 in / 4 out per group), pad=1.
// in/out layout [n][c=32][s]; w: [32][4][3][3][3]
// ---------------------------------------------------------------------------
__global__ void gconv3_kernel(const float* __restrict__ in,
                              const float* __restrict__ w,
                              const float* __restrict__ b,
                              float* __restrict__ out)
{
    const int i = blockIdx.x * blockDim.x + threadIdx.x;
    const int total = BATCH * 32 * SPAT;
    if (i >= total) return;
    const int s = i % SPAT;
    const int o = (i / SPAT) % 32;
    const int n = i / (SPAT * 32);
    const int xw = s % W_OUT;
    const int yw = (s / W_OUT) % H_OUT;
    const int zw = s / (W_OUT * H_OUT);
    const int g = o >> 2;
    float acc = b[o];
    for (int ci = 0; ci < 4; ++ci) {
        const float* ip = in + (long)(n * 32 + g * 4 + ci) * SPAT;
        const float* wp = w + ((long)o * 4 + ci) * 27;
        for (int dz = 0; dz < 3; ++dz) {
            const int zz = zw + dz - 1;
            if ((unsigned)zz >= D_OUT) continue;
            for (int dy = 0; dy < 3; ++dy) {
                const int yy = yw + dy - 1;
                if ((unsigned)yy >= H_OUT) continue;
                for (int dx = 0; dx < 3; ++dx) {
                    const int xx = xw + dx - 1;
                    if ((unsigned)xx >= W_OUT) continue;
                    acc = fmaf(ip[(zz * H_OUT + yy) * W_OUT + xx],
                               wp[(dz * 3 + dy) * 3 + dx], acc);
                }
            }
        }
    }
    out[i] = acc;
}

// ---------------------------------------------------------------------------
// LayerNorm over channels, transposing [n][c][s] -> token-major [n*s][c].
// ---------------------------------------------------------------------------
__global__ void layernorm_tok_kernel(const float* __restrict__ t3,
                                     const float* __restrict__ lw,
                                     const float* __restrict__ lb,
                                     float* __restrict__ tok)
{
    const int i = blockIdx.x * blockDim.x + threadIdx.x;
    if (i >= NTOK) return;
    const int n = i / SPAT;
    const int s = i % SPAT;
    const float* p = t3 + (long)n * 32 * SPAT + s;
    float m = 0.0f;
#pragma unroll
    for (int c = 0; c < CMODEL; ++c) m += p[(long)c * SPAT];
    m *= (1.0f / CMODEL);
    float v = 0.0f;
#pragma unroll
    for (int c = 0; c < CMODEL; ++c) {
        const float d = p[(long)c * SPAT] - m;
        v = fmaf(d, d, v);
    }
    const float inv = rsqrtf(v * (1.0f / CMODEL) + 1e-5f);
    float* o = tok + (long)i * CMODEL;
#pragma unroll
    for (int c = 0; c < CMODEL; ++c)
        o[c] = (p[(long)c * SPAT] - m) * inv * lw[c] + lb[c];
}

// ---------------------------------------------------------------------------
// Causal depthwise conv1d (k=4) over sequence + SiLU.
// xz: [n*l][128] (xi = cols 0..63), out xia: [n*l][64]
// ---------------------------------------------------------------------------
__global__ void conv1d_silu_kernel(const float* __restrict__ xz,
                                   const float* __restrict__ cw,
                                   const float* __restrict__ cb,
                                   float* __restrict__ xia)
{
    const int i = blockIdx.x * blockDim.x + threadIdx.x;
    const int total = NTOK * DINNER;
    if (i >= total) return;
    const int d = i % DINNER;
    const int l = (i / DINNER) % SPAT;
    const int n = i / (DINNER * SPAT);
    float acc = cb[d];
#pragma unroll
    for (int j = 0; j < 4; ++j) {
        const int ll = l - 3 + j;
        if (ll >= 0)
            acc = fmaf(xz[((long)n * SPAT + ll) * 128 + d], cw[d * 4 + j], acc);
    }
    xia[i] = silu_f(acc);
}

// ---------------------------------------------------------------------------
// dt = softplus(dt_pre @ dt_proj_w^T + dt_proj_b)   (K=2, tiny -> scalar)
// ---------------------------------------------------------------------------
__global__ void dt_softplus_kernel(const float* __restrict__ xdbl,
                                   const float* __restrict__ dtw,
                                   const float* __restrict__ dtb,
                                   float* __restrict__ dt)
{
    const int i = blockIdx.x * blockDim.x + threadIdx.x;
    const int total = NTOK * DINNER;
    if (i >= total) return;
    const int d = i % DINNER;
    const long t = i / DINNER;
    const float v = fmaf(xdbl[t * XPROJ_N + 0], dtw[d * DTRANK + 0],
                    fmaf(xdbl[t * XPROJ_N + 1], dtw[d * DTRANK + 1], dtb[d]));
    dt[i] = softplus_f(v);
}

// ---------------------------------------------------------------------------
// Chunked parallel selective scan.
// Recurrence h <- e*h + f with e = exp(dt*A), f = dt*B*u is associative:
//   (e2,f2) o (e1,f1) = (e2*e1, e2*f1 + f2)
// Phase 1: per (b,d,chunk): local scan from h=0 -> per-state (P=prod e, Q=h_end)
// Phase 2: per (b,d,n): serial combine over NCHUNK chunk summaries -> H0 per chunk
// Phase 3: per (b,d,chunk): re-run chunk seeded with H0, emit y = sum_n h*C + u*D
// Serial critical path drops 16384 -> 3*128 dependent steps.
// Summary layout: [b][chunk][d][n]
// ---------------------------------------------------------------------------
__global__ __launch_bounds__(256) void scan_phase1_kernel(
    const float* __restrict__ xia, const float* __restrict__ dt,
    const float* __restrict__ xdbl, const float* __restrict__ A_log,
    float* __restrict__ P, float* __restrict__ Q)
{
    const int i = blockIdx.x * blockDim.x + threadIdx.x;
    const int total = BATCH * DINNER * NCHUNK;
    if (i >= total) return;
    const int c = i % NCHUNK;
    const int d = (i / NCHUNK) % DINNER;
    const int b = i / (NCHUNK * DINNER);

    float A[DSTATE], p[DSTATE], q[DSTATE];
#pragma unroll
    for (int j = 0; j < DSTATE; ++j) {
        A[j] = -__expf(A_log[d * DSTATE + j]);
        p[j] = 1.0f;
        q[j] = 0.0f;
    }
    const long t0 = (long)b * SPAT + (long)c * CLEN;
    for (int l = 0; l < CLEN; ++l) {
        const long t = t0 + l;
        const float dtv = dt[t * DINNER + d];
        const float u   = xia[t * DINNER + d];
        const float dtu = dtv * u;
        const float* bc = xdbl + t * XPROJ_N;
#pragma unroll
        for (int j = 0; j < DSTATE; ++j) {
            const float e = __expf(dtv * A[j]);
            p[j] *= e;
            q[j] = fmaf(e, q[j], dtu * bc[DTRANK + j]);
        }
    }
    const long o = (((long)b * NCHUNK + c) * DINNER + d) * DSTATE;
#pragma unroll
    for (int j = 0; j < DSTATE; ++j) { P[o + j] = p[j]; Q[o + j] = q[j]; }
}

__global__ __launch_bounds__(256) void scan_phase2_kernel(
    const float* __restrict__ P, const float* __restrict__ Q,
    float* __restrict__ H0)
{
    const int i = blockIdx.x * blockDim.x + threadIdx.x;
    const int total = BATCH * DINNER * DSTATE;   // 2048
    if (i >= total) return;
    const int n = i % DSTATE;
    const int d = (i / DSTATE) % DINNER;
    const int b = i / (DSTATE * DINNER);
    const long stride = (long)DINNER * DSTATE;
    long o = ((long)b * NCHUNK * DINNER + d) * DSTATE + n;
    float H = 0.0f;
    for (int c = 0; c < NCHUNK; ++c) {
        H0[o] = H;
        H = fmaf(P[o], H, Q[o]);
        o += stride;
    }
}

__global__ __launch_bounds__(256) void scan_phase3_kernel(
    const float* __restrict__ xia, const float* __restrict__ dt,
    const float* __restrict__ xdbl, const float* __restrict__ A_log,
    const float* __restrict__ Dp, const float* __restrict__ H0,
    float* __restrict__ ys)
{
    const int i = blockIdx.x * blockDim.x + threadIdx.x;
    const int total = BATCH * DINNER * NCHUNK;
    if (i >= total) return;
    const int c = i % NCHUNK;
    const int d = (i / NCHUNK) % DINNER;
    const int b = i / (NCHUNK * DINNER);

    float A[DSTATE], h[DSTATE];
    const long o = (((long)b * NCHUNK + c) * DINNER + d) * DSTATE;
#pragma unroll
    for (int j = 0; j < DSTATE; ++j) {
        A[j] = -__expf(A_log[d * DSTATE + j]);
        h[j] = H0[o + j];
    }
    const float Dd = Dp[d];
    const long t0 = (long)b * SPAT + (long)c * CLEN;
    for (int l = 0; l < CLEN; ++l) {
        const long t = t0 + l;
        const float dtv = dt[t * DINNER + d];
        const float u   = xia[t * DINNER + d];
        const float dtu = dtv * u;
        const float* bc = xdbl + t * XPROJ_N;
        float y = 0.0f;
#pragma unroll
        for (int j = 0; j < DSTATE; ++j) {
            const float e = __expf(dtv * A[j]);
            h[j] = fmaf(e, h[j], dtu * bc[DTRANK + j]);
            y = fmaf(h[j], bc[DTRANK + DSTATE + j], y);
        }
        ys[t * DINNER + d] = fmaf(u, Dd, y);
    }
}

// ---------------------------------------------------------------------------
// y *= silu(z)
// ---------------------------------------------------------------------------
__global__ void gate_kernel(float* __restrict__ ys, const float* __restrict__ xz)
{
    const int i = blockIdx.x * blockDim.x + threadIdx.x;
    const int total = NTOK * DINNER;
    if (i >= total) return;
    const int d = i % DINNER;
    const long t = i / DINNER;
    ys[i] *= silu_f(xz[t * 128 + DINNER + d]);
}

// ---------------------------------------------------------------------------
extern "C" void kernel_launch(void* const* d_in, const int* in_sizes, int n_in,
                              void* d_out, int out_size, void* d_ws, size_t ws_size,
                              hipStream_t stream) {
    (void)in_sizes; (void)n_in; (void)out_size; (void)ws_size;
    const float* x1        = (const float*)d_in[0];
    const float* x2        = (const float*)d_in[1];
    const float* up_w      = (const float*)d_in[2];
    const float* up_b      = (const float*)d_in[3];
    const float* dc1_w     = (const float*)d_in[4];
    const float* dc1_b     = (const float*)d_in[5];
    const float* dc2_w     = (const float*)d_in[6];
    const float* dc2_b     = (const float*)d_in[7];
    const float* dc3_w     = (const float*)d_in[8];
    const float* dc3_b     = (const float*)d_in[9];
    const float* ln_w      = (const float*)d_in[10];
    const float* ln_b      = (const float*)d_in[11];
    const float* in_proj_w = (const float*)d_in[12];
    const float* conv1d_w  = (const float*)d_in[13];
    const float* conv1d_b  = (const float*)d_in[14];
    const float* x_proj_w  = (const float*)d_in[15];
    const float* dt_proj_w = (const float*)d_in[16];
    const float* dt_proj_b = (const float*)d_in[17];
    const float* A_log     = (const float*)d_in[18];
    const float* D_param   = (const float*)d_in[19];
    const float* out_proj_w= (const float*)d_in[20];
    float* out = (float*)d_out;
    float* ws  = (float*)d_ws;

    float* xcat = ws + WS_XCAT;
    float* t1   = ws + WS_T1;
    float* t2   = ws + WS_T2;
    float* t3   = ws + WS_T3;
    float* tok  = ws + WS_TOK;
    float* xz   = ws + WS_XZ;
    float* xia  = ws + WS_XIA;
    float* xdbl = ws + WS_XDBL;
    float* Psum = ws + WS_P;
    float* Qsum = ws + WS_Q;
    float* H0   = ws + WS_H0;
    float* dt   = ws + WS_DT;
    float* ys   = ws + WS_YS;

    // 1. upsample + concat -> xcat [2][64][16384]
    upsample_concat_kernel<<<(BATCH * 64 * SPAT + 255) / 256, 256, 0, stream>>>(
        x1, x2, up_w, up_b, xcat);

    // 2. dc1: 1x1x1 conv 64->32 as channel-major WMMA GEMM -> t1
    gemm_wmma_f32<<<dim3(SPAT / 16, 32 / 16, BATCH), 32, 0, stream>>>(
        xcat, dc1_w, dc1_b, t1, SPAT, 32, 64,
        (long)64 * SPAT, 1L, (long)SPAT,
        (long)32 * SPAT, 1L, (long)SPAT);
    instnorm_leaky_kernel<<<BATCH * 32, 256, 0, stream>>>(t1);

    // 3. dc2: grouped 3x3x3 conv -> t2
    gconv3_kernel<<<(BATCH * 32 * SPAT + 255) / 256, 256, 0, stream>>>(
        t1, dc2_w, dc2_b, t2);
    instnorm_leaky_kernel<<<BATCH * 32, 256, 0, stream>>>(t2);

    // 4. dc3: 1x1x1 conv 32->32 WMMA GEMM -> t3
    gemm_wmma_f32<<<dim3(SPAT / 16, 32 / 16, BATCH), 32, 0, stream>>>(
        t2, dc3_w, dc3_b, t3, SPAT, 32, 32,
        (long)32 * SPAT, 1L, (long)SPAT,
        (long)32 * SPAT, 1L, (long)SPAT);
    instnorm_leaky_kernel<<<BATCH * 32, 256, 0, stream>>>(t3);

    // 5. LayerNorm + transpose to token-major -> tok [32768][32]
    layernorm_tok_kernel<<<(NTOK + 255) / 256, 256, 0, stream>>>(t3, ln_w, ln_b, tok);

    // 6. in_proj: [32768,32]x[32,128] WMMA GEMM -> xz [32768][128]
    gemm_wmma_f32<<<dim3(NTOK / 16, 128 / 16, 1), 32, 0, stream>>>(
        tok, in_proj_w, nullptr, xz, NTOK, 128, 32,
        0L, (long)CMODEL, 1L,
        0L, 128L, 1L);

    // 7. causal depthwise conv1d + silu -> xia [32768][64]
    conv1d_silu_kernel<<<(NTOK * DINNER + 255) / 256, 256, 0, stream>>>(
        xz, conv1d_w, conv1d_b, xia);

    // 8. x_proj: [32768,64]x[64,34] WMMA GEMM -> xdbl [32768][34]
    gemm_wmma_f32<<<dim3(NTOK / 16, (XPROJ_N + 15) / 16, 1), 32, 0, stream>>>(
        xia, x_proj_w, nullptr, xdbl, NTOK, XPROJ_N, DINNER,
        0L, (long)DINNER, 1L,
        0L, (long)XPROJ_N, 1L);

    // 9. dt = softplus(dt_pre @ dt_proj_w^T + b) -> dt [32768][64]
    dt_softplus_kernel<<<(NTOK * DINNER + 255) / 256, 256, 0, stream>>>(
        xdbl, dt_proj_w, dt_proj_b, dt);

    // 10. chunked parallel selective scan -> ys [32768][64]
    scan_phase1_kernel<<<(BATCH * DINNER * NCHUNK + 255) / 256, 256, 0, stream>>>(
        xia, dt, xdbl, A_log, Psum, Qsum);
    scan_phase2_kernel<<<(BATCH * DINNER * DSTATE + 255) / 256, 256, 0, stream>>>(
        Psum, Qsum, H0);
    scan_phase3_kernel<<<(BATCH * DINNER * NCHUNK + 255) / 256, 256, 0, stream>>>(
        xia, dt, xdbl, A_log, D_param, H0, ys);

    // 11. y *= silu(z)
    gate_kernel<<<(NTOK * DINNER + 255) / 256, 256, 0, stream>>>(ys, xz);

    // 12. out_proj: [16384,64]x[64,32] per batch, strided store into [b][c][s]
    gemm_wmma_f32<<<dim3(SPAT / 16, 32 / 16, BATCH), 32, 0, stream>>>(
        ys, out_proj_w, nullptr, out, SPAT, 32, DINNER,
        (long)SPAT * DINNER, (long)DINNER, 1L,
        (long)32 * SPAT, 1L, (long)SPAT);
}